// ProteinGCN_28355374088744
// MI455X (gfx1250) — compile-verified
//
#include <hip/hip_runtime.h>

typedef __attribute__((ext_vector_type(2))) float v2f;
typedef __attribute__((ext_vector_type(8))) float v8f;

// ---------------- utility kernels ----------------

__global__ void fill_const_kernel(float* __restrict__ p, float v, int nElem) {
    int i = blockIdx.x * blockDim.x + threadIdx.x;
    if (i < nElem) p[i] = v;
}

__global__ void deg_count_kernel(const int* __restrict__ col, float* __restrict__ deg, int E) {
    int e = blockIdx.x * blockDim.x + threadIdx.x;
    if (e < E) atomicAdd(&deg[col[e]], 1.0f);
}

__global__ void rsqrt_inplace_kernel(float* __restrict__ d, int n) {
    int i = blockIdx.x * blockDim.x + threadIdx.x;
    if (i < n) d[i] = rsqrtf(d[i]);   // deg >= 1 always (self loop)
}

__global__ void edge_norm_kernel(const int* __restrict__ row, const int* __restrict__ col,
                                 const float* __restrict__ dinv, float* __restrict__ norm, int E) {
    int e = blockIdx.x * blockDim.x + threadIdx.x;
    if (e < E) norm[e] = dinv[row[e]] * dinv[col[e]];
}

// ---------------- WMMA fp32 GEMM:  C[M x (NT*16)] = A[M x K] @ B[K x (NT*16)] ----------------
// One wave computes a 16 x (NT*16) stripe using V_WMMA_F32_16X16X4_F32.
// fp32 A (16x4) layout : lane L holds row M=L&15; VGPR0/1 = K = 2*(L>>4), 2*(L>>4)+1
// fp32 B (4x16) layout : lane L holds col N=L&15; VGPR0/1 = K = 2*(L>>4), 2*(L>>4)+1
// fp32 C (16x16) layout: lane L holds col N=L&15; VGPR i = row M = i + 8*(L>>4)
template <int K, int NT>
__global__ void wmma_gemm_kernel(const float* __restrict__ A, const float* __restrict__ B,
                                 float* __restrict__ C, int M) {
    const int NCOL  = NT * 16;
    const int lane  = threadIdx.x & 31;
    const int wave  = threadIdx.x >> 5;
    const int mtile = blockIdx.x * (blockDim.x >> 5) + wave;
    if (mtile * 16 >= M) return;                 // wave-uniform exit: EXEC stays all-ones

    const int mrow0 = mtile * 16;
    const int sel   = lane & 15;                 // A: row-in-tile, B/C: col-in-tile
    const int khalf = (lane >> 4) * 2;           // 0 or 2

    int arow = mrow0 + sel;
    if (arow >= M) arow = M - 1;                 // clamped read keeps EXEC full

    v8f acc[NT] = {};

    for (int k = 0; k < K; k += 4) {
        const v2f a = *(const v2f*)(A + (size_t)arow * K + k + khalf);
#pragma unroll
        for (int t = 0; t < NT; ++t) {
            v2f b;
            b.x = B[(size_t)(k + khalf)     * NCOL + t * 16 + sel];
            b.y = B[(size_t)(k + khalf + 1) * NCOL + t * 16 + sel];
            acc[t] = __builtin_amdgcn_wmma_f32_16x16x4_f32(
                /*neg_a=*/false, a, /*neg_b=*/false, b,
                /*c_mod=*/(short)0, acc[t], /*reuse_a=*/false, /*reuse_b=*/false);
        }
    }

#pragma unroll
    for (int t = 0; t < NT; ++t) {
#pragma unroll
        for (int i = 0; i < 8; ++i) {
            int m = mrow0 + i + 8 * (lane >> 4);
            if (m < M) C[(size_t)m * NCOL + t * 16 + sel] = acc[t][i];
        }
    }
}

// ---------------- edge scatter:  out[col] += h[row] * norm  (float4 per thread) ----------------
template <int F4>   // F4 = F/4
__global__ void edge_scatter_kernel(const int* __restrict__ row, const int* __restrict__ col,
                                    const float* __restrict__ norm, const float* __restrict__ h,
                                    float* __restrict__ out, int E) {
    int idx = blockIdx.x * blockDim.x + threadIdx.x;
    if (idx >= E * F4) return;
    const int e  = idx / F4;
    const int f4 = (idx % F4) * 4;
    const int F  = F4 * 4;
    const int r  = row[e];
    const int c  = col[e];
    const float w = norm[e];
    const float4 hv = *(const float4*)(h + (size_t)r * F + f4);
    float* o = out + (size_t)c * F + f4;
    atomicAdd(o + 0, hv.x * w);
    atomicAdd(o + 1, hv.y * w);
    atomicAdd(o + 2, hv.z * w);
    atomicAdd(o + 3, hv.w * w);
}

// ---------------- finalize: out = relu(out + h * dinv^2 + b) ----------------
__global__ void finalize_kernel(float* __restrict__ out, const float* __restrict__ h,
                                const float* __restrict__ dinv, const float* __restrict__ b,
                                int n, int F) {
    int i = blockIdx.x * blockDim.x + threadIdx.x;
    if (i >= n * F) return;
    const int v = i / F;
    const int f = i - v * F;
    const float d = dinv[v];
    const float val = out[i] + h[i] * d * d + b[f];
    out[i] = val > 0.0f ? val : 0.0f;
}

// ---------------- FC (32 -> 3) + log_softmax + embedding copy ----------------
__global__ void fc_logsoftmax_kernel(const float* __restrict__ emb, const float* __restrict__ Wfc,
                                     const float* __restrict__ bfc, float* __restrict__ out_ls,
                                     float* __restrict__ out_emb, int n) {
    int v = blockIdx.x * blockDim.x + threadIdx.x;
    if (v >= n) return;
    const float* ev = emb + (size_t)v * 32;
    float o0 = bfc[0], o1 = bfc[1], o2 = bfc[2];
#pragma unroll
    for (int k = 0; k < 32; ++k) {
        const float x = ev[k];
        o0 = fmaf(x, Wfc[k * 3 + 0], o0);
        o1 = fmaf(x, Wfc[k * 3 + 1], o1);
        o2 = fmaf(x, Wfc[k * 3 + 2], o2);
    }
    const float m   = fmaxf(o0, fmaxf(o1, o2));
    const float lse = m + logf(expf(o0 - m) + expf(o1 - m) + expf(o2 - m));
    out_ls[(size_t)v * 3 + 0] = o0 - lse;
    out_ls[(size_t)v * 3 + 1] = o1 - lse;
    out_ls[(size_t)v * 3 + 2] = o2 - lse;
#pragma unroll
    for (int k = 0; k < 32; ++k) out_emb[(size_t)v * 32 + k] = ev[k];
}

// ---------------- host-side orchestration ----------------

static inline int cdiv(int a, int b) { return (a + b - 1) / b; }

extern "C" void kernel_launch(void* const* d_in, const int* in_sizes, int n_in,
                              void* d_out, int out_size, void* d_ws, size_t ws_size,
                              hipStream_t stream) {
    const float* x   = (const float*)d_in[0];
    const int*  eidx = (const int*)d_in[1];
    const float* W1 = (const float*)d_in[2]; const float* b1 = (const float*)d_in[3];
    const float* W2 = (const float*)d_in[4]; const float* b2 = (const float*)d_in[5];
    const float* W3 = (const float*)d_in[6]; const float* b3 = (const float*)d_in[7];
    const float* Wfc = (const float*)d_in[8]; const float* bfc = (const float*)d_in[9];

    const int F0 = 128, F1 = 128, F2 = 64, F3 = 32;
    const int n = in_sizes[0] / F0;
    const int E = in_sizes[1] / 2;
    const int* row = eidx;       // sources
    const int* col = eidx + E;   // targets

    // workspace layout (floats)
    float* ws   = (float*)d_ws;
    float* dinv = ws;                            // n
    float* norm = dinv + n;                      // E
    float* bufA = norm + E;                      // n * 128 (gemm output)
    float* bufB = bufA + (size_t)n * 128;        // n * 128 (aggregate / layer output)

    const int T = 256;

    // ---- degree -> dinv -> per-edge norm (reused by all 3 layers) ----
    fill_const_kernel<<<cdiv(n, T), T, 0, stream>>>(dinv, 1.0f, n);   // self-loop contributes 1
    deg_count_kernel<<<cdiv(E, T), T, 0, stream>>>(col, dinv, E);
    rsqrt_inplace_kernel<<<cdiv(n, T), T, 0, stream>>>(dinv, n);
    edge_norm_kernel<<<cdiv(E, T), T, 0, stream>>>(row, col, dinv, norm, E);

    const int mtiles = cdiv(n, 16);
    const int gblk   = cdiv(mtiles, 4);          // 4 waves / block, 1 wave per 16-row stripe

    // ---- layer 1: 128 -> 128 ----
    wmma_gemm_kernel<128, 8><<<gblk, 128, 0, stream>>>(x, W1, bufA, n);
    fill_const_kernel<<<cdiv(n * F1, T), T, 0, stream>>>(bufB, 0.0f, n * F1);
    edge_scatter_kernel<32><<<cdiv(E * 32, T), T, 0, stream>>>(row, col, norm, bufA, bufB, E);
    finalize_kernel<<<cdiv(n * F1, T), T, 0, stream>>>(bufB, bufA, dinv, b1, n, F1);

    // ---- layer 2: 128 -> 64 ----
    wmma_gemm_kernel<128, 4><<<gblk, 128, 0, stream>>>(bufB, W2, bufA, n);
    fill_const_kernel<<<cdiv(n * F2, T), T, 0, stream>>>(bufB, 0.0f, n * F2);
    edge_scatter_kernel<16><<<cdiv(E * 16, T), T, 0, stream>>>(row, col, norm, bufA, bufB, E);
    finalize_kernel<<<cdiv(n * F2, T), T, 0, stream>>>(bufB, bufA, dinv, b2, n, F2);

    // ---- layer 3: 64 -> 32 (embeddings) ----
    wmma_gemm_kernel<64, 2><<<gblk, 128, 0, stream>>>(bufB, W3, bufA, n);
    fill_const_kernel<<<cdiv(n * F3, T), T, 0, stream>>>(bufB, 0.0f, n * F3);
    edge_scatter_kernel<8><<<cdiv(E * 8, T), T, 0, stream>>>(row, col, norm, bufA, bufB, E);
    finalize_kernel<<<cdiv(n * F3, T), T, 0, stream>>>(bufB, bufA, dinv, b3, n, F3);

    // ---- FC + log_softmax; outputs: [log_softmax (n*3)] then [embeddings (n*32)] ----
    float* out_ls  = (float*)d_out;
    float* out_emb = out_ls + (size_t)n * 3;
    fc_logsoftmax_kernel<<<cdiv(n, T), T, 0, stream>>>(bufB, Wfc, bfc, out_ls, out_emb, n);
}